// GATLayer_1735166787955
// MI455X (gfx1250) — compile-verified
//
#include <hip/hip_runtime.h>
#include <hip/hip_bf16.h>
#include <math.h>

// GAT layer on MI455X (gfx1250, wave32):
//   h_in = x@lin_w+b (residual), xp = x@gat_w  -> WMMA f32 16x16x4 dual GEMM
//   per-head edge softmax (+self loops) via 2 atomic passes
//   scatter-add aggregation, BatchNorm(batch stats), ELU, residual.
// N=100000 (multiple of 16), E=1600000, IN=128, H=4, C=32, D=128.

#define DD 128
#define HH 4
#define CC 32

typedef float v2f __attribute__((ext_vector_type(2)));
typedef float v8f __attribute__((ext_vector_type(8)));
typedef int v4i __attribute__((ext_vector_type(4)));

#if defined(__AMDGCN__) && __has_builtin(__builtin_amdgcn_global_load_async_to_lds_b128)
#define HAVE_ASYNC_LDS 1
#else
#define HAVE_ASYNC_LDS 0
#endif

__device__ __forceinline__ void wait_async_zero() {
#if defined(__AMDGCN__)
#if __has_builtin(__builtin_amdgcn_s_wait_asynccnt)
  __builtin_amdgcn_s_wait_asynccnt(0);
#else
  asm volatile("s_wait_asynccnt 0" ::: "memory");
#endif
#endif
}

// ---------------------------------------------------------------- utilities
__device__ __forceinline__ void atomicMaxFloat(float* addr, float val) {
  // Monotone float max via integer encodings (init with -inf):
  if (val >= 0.0f) {
    atomicMax((int*)addr, __float_as_int(val));
  } else {
    atomicMin((unsigned int*)addr, (unsigned int)__float_as_int(val));
  }
}

// ---------------------------------------------------------------- 0) init
__global__ void init_ws_kernel(float* __restrict__ mbuf, float* __restrict__ denom,
                               float* __restrict__ out_acc, float* __restrict__ col_sum,
                               float* __restrict__ col_sumsq, int N) {
  int t = blockIdx.x * blockDim.x + threadIdx.x;
  if (t < N * DD) out_acc[t] = 0.0f;
  if (t < N * HH) { mbuf[t] = -INFINITY; denom[t] = 0.0f; }
  if (t < DD) { col_sum[t] = 0.0f; col_sumsq[t] = 0.0f; }
}

// ---------------------------------------------------------------- 0b) weight pre-pack
// wp[(k>>1)*256 + col*2 + (k&1)] = w[k*128 + col]
// -> each WMMA B fragment (rows ka, ka+1 at column col) is one coalesced b64.
__global__ void pack_w_kernel(const float* __restrict__ w, float* __restrict__ wp) {
  int t = blockIdx.x * blockDim.x + threadIdx.x;  // 16384
  int k = t >> 7;
  int col = t & (DD - 1);
  wp[(k >> 1) * 256 + col * 2 + (k & 1)] = w[t];
}

// ---------------------------------------------------------------- 1) dual GEMM (WMMA f32, LDS-staged A)
// One 256-thread block (8 waves) per 16-row tile; wave w owns column tile w.
// x-tile (16x128) staged into LDS via GLOBAL_LOAD_ASYNC_TO_LDS_B128 (fallback:
// manual b128 copy), A fed from ds_load, B from packed b64 loads.
__global__ void __launch_bounds__(256)
gemm_dual_wmma(const float* __restrict__ x, const float* __restrict__ lin_wp,
               const float* __restrict__ lin_b, const float* __restrict__ gat_wp,
               float* __restrict__ h_in, float* __restrict__ xp, int N) {
  __shared__ float xs[16 * DD];  // 8 KB

  const int tid = threadIdx.x;
  const int rowTile = blockIdx.x;

  // cooperative stage: 2048 floats = 512 x b128 = 256 threads x 2
  {
    const float4* src4 = (const float4*)(x + (size_t)rowTile * 16 * DD);
    float4* dst4 = (float4*)xs;
#if HAVE_ASYNC_LDS
    __builtin_amdgcn_global_load_async_to_lds_b128(
        (__attribute__((address_space(1))) v4i*)(src4 + tid),
        (__attribute__((address_space(3))) v4i*)(dst4 + tid), 0, 0);
    __builtin_amdgcn_global_load_async_to_lds_b128(
        (__attribute__((address_space(1))) v4i*)(src4 + tid + 256),
        (__attribute__((address_space(3))) v4i*)(dst4 + tid + 256), 0, 0);
    wait_async_zero();
#else
    dst4[tid] = src4[tid];
    dst4[tid + 256] = src4[tid + 256];
#endif
  }
  __syncthreads();

  const int wave = tid >> 5;       // 0..7 == column tile
  const int lane = tid & 31;
  const int l16 = lane & 15;
  const int half = lane >> 4;      // 0: K=k0,k0+1 ; 1: K=k0+2,k0+3 (ISA A layout)
  const int col = wave * 16 + l16; // 0..127
  const float* __restrict__ xrow = xs + l16 * DD;
  const float* __restrict__ blp = lin_wp + col * 2;
  const float* __restrict__ bgp = gat_wp + col * 2;

  v8f acc_lin = {};
  v8f acc_gat = {};
#pragma unroll 8
  for (int k0 = 0; k0 < DD; k0 += 4) {
    const int ka = k0 + half * 2;
    const int p = (k0 >> 1) + half;  // K-pair index
    v2f a;
    a.x = xrow[ka];                  // ds_load (2addr b64 pairs)
    a.y = xrow[ka + 1];
    v2f bl = *(const v2f*)(blp + p * 256);  // one b64 per fragment
    v2f bg = *(const v2f*)(bgp + p * 256);
    acc_lin = __builtin_amdgcn_wmma_f32_16x16x4_f32(false, a, false, bl,
                                                    (short)0, acc_lin, false, false);
    acc_gat = __builtin_amdgcn_wmma_f32_16x16x4_f32(false, a, false, bg,
                                                    (short)0, acc_gat, false, false);
  }

  // C/D layout: VGPR i holds rows M=i (lanes 0-15) / M=i+8 (lanes 16-31).
  const float bias = lin_b[col];
  const size_t rbase = (size_t)rowTile * 16 + half * 8;  // N % 16 == 0: no guards
#pragma unroll
  for (int i = 0; i < 8; ++i) {
    h_in[(rbase + i) * DD + col] = acc_lin[i] + bias;
    xp[(rbase + i) * DD + col] = acc_gat[i];
  }
}

// ---------------------------------------------------------------- 2) attention logits
// One wave per node: lane owns channels 4*lane..4*lane+3 (head = lane>>3),
// 8-lane shfl_xor reduction per head.
__global__ void attn_logits_kernel(const float* __restrict__ xp,
                                   const float* __restrict__ att_src,
                                   const float* __restrict__ att_dst,
                                   float* __restrict__ a_src, float* __restrict__ a_dst,
                                   int N) {
  int gt = blockIdx.x * blockDim.x + threadIdx.x;
  int w = gt >> 5;
  int lane = gt & 31;
  if (w >= N) return;
  float4 xv = *(const float4*)(xp + (size_t)w * DD + lane * 4);
  float4 s4 = *(const float4*)(att_src + lane * 4);  // att[h][c], h=lane>>3
  float4 d4 = *(const float4*)(att_dst + lane * 4);
  float ss = xv.x * s4.x + xv.y * s4.y + xv.z * s4.z + xv.w * s4.w;
  float ds = xv.x * d4.x + xv.y * d4.y + xv.z * d4.z + xv.w * d4.w;
  ss += __shfl_xor(ss, 1); ds += __shfl_xor(ds, 1);
  ss += __shfl_xor(ss, 2); ds += __shfl_xor(ds, 2);
  ss += __shfl_xor(ss, 4); ds += __shfl_xor(ds, 4);
  if ((lane & 7) == 0) {
    int h = lane >> 3;
    a_src[w * HH + h] = ss;
    a_dst[w * HH + h] = ds;
  }
}

// ---------------------------------------------------------------- 3) per-dst max (softmax pass 1)
__global__ void edge_max_kernel(const int* __restrict__ src, const int* __restrict__ dst,
                                const float* __restrict__ a_src, const float* __restrict__ a_dst,
                                float* __restrict__ mbuf, int E, int N) {
  int t = blockIdx.x * blockDim.x + threadIdx.x;
  if (t >= E + N) return;
  int s, d;
  if (t < E) { s = src[t]; d = dst[t]; } else { s = d = t - E; }  // self loops
  float as[4], ad[4];
  *(float4*)as = *(const float4*)(a_src + 4 * (size_t)s);
  *(float4*)ad = *(const float4*)(a_dst + 4 * (size_t)d);
#pragma unroll
  for (int h = 0; h < HH; ++h) {
    float e = as[h] + ad[h];
    e = (e > 0.0f) ? e : 0.2f * e;  // leaky_relu(0.2)
    atomicMaxFloat(&mbuf[d * HH + h], e);
  }
}

// ---------------------------------------------------------------- 4) fused exp/denom/weighted scatter-add
// One wave32 per edge; lane owns 4 consecutive channels (one b128 gather).
__global__ void edge_accum_kernel(const int* __restrict__ src, const int* __restrict__ dst,
                                  const float* __restrict__ a_src, const float* __restrict__ a_dst,
                                  const float* __restrict__ mbuf, const float* __restrict__ xp,
                                  float* __restrict__ denom, float* __restrict__ out_acc,
                                  int E, int N) {
  int gt = blockIdx.x * blockDim.x + threadIdx.x;
  int w = gt >> 5;
  int lane = gt & 31;
  if (w >= E + N) return;
  int s, d;
  if (w < E) { s = src[w]; d = dst[w]; } else { s = d = w - E; }
  float as[4], ad[4], mm[4], ex[4];
  *(float4*)as = *(const float4*)(a_src + 4 * (size_t)s);
  *(float4*)ad = *(const float4*)(a_dst + 4 * (size_t)d);
  *(float4*)mm = *(const float4*)(mbuf + 4 * (size_t)d);
#pragma unroll
  for (int h = 0; h < HH; ++h) {
    float e = as[h] + ad[h];
    e = (e > 0.0f) ? e : 0.2f * e;
    ex[h] = __expf(e - mm[h]);
  }
  // channels 4*lane..4*lane+3 all belong to head lane>>3
  float xv[4];
  *(float4*)xv = *(const float4*)(xp + (size_t)s * DD + lane * 4);
  float* outd = out_acc + (size_t)d * DD + lane * 4;
  const float exh = ex[lane >> 3];
  atomicAdd(outd + 0, exh * xv[0]);
  atomicAdd(outd + 1, exh * xv[1]);
  atomicAdd(outd + 2, exh * xv[2]);
  atomicAdd(outd + 3, exh * xv[3]);
  if (lane < HH) atomicAdd(&denom[(size_t)d * HH + lane], ex[lane]);
}

// ---------------------------------------------------------------- 5) per-node finalize + BN column stats
__global__ void finalize_stats_kernel(float* __restrict__ out_acc,  // in-place -> h
                                      const float* __restrict__ denom,
                                      const float* __restrict__ gat_bias,
                                      float* __restrict__ col_sum, float* __restrict__ col_sumsq,
                                      int N) {
  const int d = threadIdx.x;  // blockDim.x == 128
  float s = 0.0f, s2 = 0.0f;
  for (int n = blockIdx.x; n < N; n += gridDim.x) {
    float v = out_acc[(size_t)n * DD + d] / denom[n * HH + (d >> 5)] + gat_bias[d];
    out_acc[(size_t)n * DD + d] = v;
    s += v;
    s2 += v * v;
  }
  atomicAdd(&col_sum[d], s);
  atomicAdd(&col_sumsq[d], s2);
}

// ---------------------------------------------------------------- 6) BN + ELU + residual
__global__ void bn_elu_residual_kernel(const float* __restrict__ h, const float* __restrict__ col_sum,
                                       const float* __restrict__ col_sumsq,
                                       const float* __restrict__ gamma, const float* __restrict__ beta,
                                       float* __restrict__ out /* holds h_in */, int N) {
  int idx = blockIdx.x * blockDim.x + threadIdx.x;
  if (idx >= N * DD) return;
  const int d = idx & (DD - 1);
  const float invN = 1.0f / (float)N;
  const float mu = col_sum[d] * invN;
  const float var = col_sumsq[d] * invN - mu * mu;  // biased variance (jnp.var default)
  float v = (h[idx] - mu) * rsqrtf(var + 1e-5f) * gamma[d] + beta[d];
  v = (v > 0.0f) ? v : (__expf(v) - 1.0f);  // ELU(alpha=1)
  out[idx] = out[idx] + v;                  // residual: h_in (already in d_out) + h
}

// ---------------------------------------------------------------- launcher
extern "C" void kernel_launch(void* const* d_in, const int* in_sizes, int n_in,
                              void* d_out, int out_size, void* d_ws, size_t ws_size,
                              hipStream_t stream) {
  const float* x        = (const float*)d_in[0];
  const int*   ei       = (const int*)d_in[1];
  const float* lin_w    = (const float*)d_in[2];
  const float* lin_b    = (const float*)d_in[3];
  const float* gat_w    = (const float*)d_in[4];
  const float* att_src  = (const float*)d_in[5];
  const float* att_dst  = (const float*)d_in[6];
  const float* gat_bias = (const float*)d_in[7];
  const float* bn_gamma = (const float*)d_in[8];
  const float* bn_beta  = (const float*)d_in[9];
  float* out = (float*)d_out;  // h_in residual buffer, then final output

  const int N = in_sizes[0] / DD;  // 100000 (multiple of 16)
  const int E = in_sizes[1] / 2;   // 1600000
  const int* e_src = ei;
  const int* e_dst = ei + E;

  // workspace layout (floats)
  float* ws        = (float*)d_ws;
  float* xp        = ws;                          // N*128
  float* out_acc   = xp + (size_t)N * DD;         // N*128
  float* a_src     = out_acc + (size_t)N * DD;    // N*4
  float* a_dst     = a_src + (size_t)N * HH;      // N*4
  float* mbuf      = a_dst + (size_t)N * HH;      // N*4
  float* denom     = mbuf + (size_t)N * HH;       // N*4
  float* col_sum   = denom + (size_t)N * HH;      // 128
  float* col_sumsq = col_sum + DD;                // 128
  float* lin_wp    = col_sumsq + DD;              // 128*128 packed weights
  float* gat_wp    = lin_wp + DD * DD;            // 128*128

  // 0) init accumulators (ws is poisoned; must re-init every call)
  init_ws_kernel<<<(N * DD + 255) / 256, 256, 0, stream>>>(mbuf, denom, out_acc,
                                                           col_sum, col_sumsq, N);
  // 0b) pack weights for b64 B-fragment loads
  pack_w_kernel<<<(DD * DD + 255) / 256, 256, 0, stream>>>(lin_w, lin_wp);
  pack_w_kernel<<<(DD * DD + 255) / 256, 256, 0, stream>>>(gat_w, gat_wp);
  // 1) dual GEMM via WMMA: h_in -> d_out, xp -> ws
  gemm_dual_wmma<<<N / 16, 256, 0, stream>>>(x, lin_wp, lin_b, gat_wp, out, xp, N);
  // 2) attention logits (1 wave per node)
  attn_logits_kernel<<<(N * 32 + 255) / 256, 256, 0, stream>>>(xp, att_src, att_dst,
                                                               a_src, a_dst, N);
  // 3) segment max
  edge_max_kernel<<<(E + N + 255) / 256, 256, 0, stream>>>(e_src, e_dst, a_src, a_dst,
                                                           mbuf, E, N);
  // 4) fused exp + denom + weighted scatter-add (1 wave per edge)
  edge_accum_kernel<<<(E + N + 7) / 8, 256, 0, stream>>>(e_src, e_dst, a_src, a_dst,
                                                         mbuf, xp, denom, out_acc, E, N);
  // 5) per-node normalize + bias, accumulate BN stats
  finalize_stats_kernel<<<256, DD, 0, stream>>>(out_acc, denom, gat_bias,
                                                col_sum, col_sumsq, N);
  // 6) BN + ELU + residual
  bn_elu_residual_kernel<<<(N * DD + 255) / 256, 256, 0, stream>>>(out_acc, col_sum, col_sumsq,
                                                                   bn_gamma, bn_beta, out, N);
}